// DroneGAT_58196806860934
// MI455X (gfx1250) — compile-verified
//
#include <hip/hip_runtime.h>
#include <math.h>

typedef __attribute__((ext_vector_type(2))) float v2f;
typedef __attribute__((ext_vector_type(8))) float v8f;

#define NEG_SLOPE 0.2f

// ---------------------------------------------------------------------------
// helpers
// ---------------------------------------------------------------------------
__device__ __forceinline__ void edge_nodes(const int* __restrict__ ei, int E, int e,
                                           int& s, int& d) {
    if (e < E) { s = ei[e]; d = ei[E + e]; }
    else       { s = e - E; d = e - E; }          // self loops appended
}

__device__ __forceinline__ void atomicMaxFloat(float* addr, float val) {
    if (val >= 0.0f)
        atomicMax((int*)addr, __float_as_int(val));
    else
        atomicMin((unsigned int*)addr, __float_as_uint(val));
}

__device__ __forceinline__ float lrelu(float v) {
    return v > 0.0f ? v : NEG_SLOPE * v;
}

// ---------------------------------------------------------------------------
// fill
// ---------------------------------------------------------------------------
__global__ void fill_kernel(float* __restrict__ p, float v, long long n) {
    long long i = (long long)blockIdx.x * blockDim.x + threadIdx.x;
    long long stride = (long long)gridDim.x * blockDim.x;
    for (; i < n; i += stride) p[i] = v;
}

// ---------------------------------------------------------------------------
// f32 WMMA GEMM:  C[M,Nf] = A[M,K] * B[K,Nf]
// One wave computes a 16x64 output slab (4 N-tiles): the A fragment is loaded
// once per K-step and fed to 4 back-to-back V_WMMA_F32_16X16X4_F32 ops (4x A
// reuse, fully-coalesced 64-float B rows).
// Requires: M%16==0, Nf%64==0, K%4==0 (true for layers 1-3)
// ---------------------------------------------------------------------------
__global__ void gemm_wmma_f32(const float* __restrict__ A, const float* __restrict__ B,
                              float* __restrict__ C, int M, int K, int Nf) {
    int wave    = blockIdx.x * (blockDim.x >> 5) + (threadIdx.x >> 5);
    int slabsN  = Nf >> 6;                 // 64-column slabs
    int tilesM  = M >> 4;
    if (wave >= tilesM * slabsN) return;
    int tm  = wave / slabsN;
    int tn4 = wave % slabsN;

    int lane = threadIdx.x & 31;
    int r    = lane & 15;                  // row (A) / col (B,C,D) within tile
    int kp   = (lane >> 4) << 1;           // lanes 0-15 -> K {0,1}; 16-31 -> K {2,3}

    v8f acc0 = {}, acc1 = {}, acc2 = {}, acc3 = {};
    const float* arow  = A + (size_t)(tm * 16 + r) * K + kp;
    const float* bbase = B + (size_t)kp * Nf + tn4 * 64 + r;

    for (int k = 0; k < K; k += 4) {
        v2f a = *(const v2f*)(arow + k);
        const float* bp = bbase + (size_t)k * Nf;
        v2f b0, b1, b2, b3;
        b0[0] = bp[0];       b0[1] = bp[Nf];
        b1[0] = bp[16];      b1[1] = bp[Nf + 16];
        b2[0] = bp[32];      b2[1] = bp[Nf + 32];
        b3[0] = bp[48];      b3[1] = bp[Nf + 48];
        acc0 = __builtin_amdgcn_wmma_f32_16x16x4_f32(false, a, false, b0, (short)0, acc0, false, false);
        acc1 = __builtin_amdgcn_wmma_f32_16x16x4_f32(false, a, false, b1, (short)0, acc1, false, false);
        acc2 = __builtin_amdgcn_wmma_f32_16x16x4_f32(false, a, false, b2, (short)0, acc2, false, false);
        acc3 = __builtin_amdgcn_wmma_f32_16x16x4_f32(false, a, false, b3, (short)0, acc3, false, false);
    }

    int rbase = tm * 16 + ((lane >> 4) << 3);   // C/D: VGPR i -> row i (+8 for hi lanes)
    int col   = tn4 * 64 + r;
    float* cp = C + (size_t)rbase * Nf + col;
#pragma unroll
    for (int i = 0; i < 8; ++i) {
        float* row = cp + (size_t)i * Nf;
        row[0]  = acc0[i];
        row[16] = acc1[i];
        row[32] = acc2[i];
        row[48] = acc3[i];
    }
}

// ---------------------------------------------------------------------------
// per-(node,head) attention logits: as = <h[n,h,:], a_src[h,:]>, ad likewise
// one wave per (node,head); C multiple of 4
// ---------------------------------------------------------------------------
__global__ void alpha_kernel(const float* __restrict__ h, const float* __restrict__ aS,
                             const float* __restrict__ aD, float* __restrict__ outS,
                             float* __restrict__ outD, int Nn, int H, int C) {
    int wid = blockIdx.x * (blockDim.x >> 5) + (threadIdx.x >> 5);
    if (wid >= Nn * H) return;
    int n  = wid / H;
    int hh = wid % H;
    int lane = threadIdx.x & 31;

    const float* hp = h  + ((size_t)n * H + hh) * C;
    const float* sp = aS + (size_t)hh * C;
    const float* dp = aD + (size_t)hh * C;

    float accs = 0.0f, accd = 0.0f;
    for (int c = lane * 4; c < C; c += 128) {
        float4 hv = *(const float4*)(hp + c);
        float4 sv = *(const float4*)(sp + c);
        float4 dv = *(const float4*)(dp + c);
        accs += hv.x * sv.x + hv.y * sv.y + hv.z * sv.z + hv.w * sv.w;
        accd += hv.x * dv.x + hv.y * dv.y + hv.z * dv.z + hv.w * dv.w;
    }
    for (int o = 16; o; o >>= 1) {
        accs += __shfl_xor(accs, o, 32);
        accd += __shfl_xor(accd, o, 32);
    }
    if (lane == 0) { outS[wid] = accs; outD[wid] = accd; }
}

// ---------------------------------------------------------------------------
// edge pass 1: segment max of leaky_relu(as[src]+ad[dst]) over dst
// ---------------------------------------------------------------------------
__global__ void edge_max_kernel(const int* __restrict__ ei, int E, int Etot,
                                const float* __restrict__ asv, const float* __restrict__ adv,
                                float* __restrict__ mx, int H) {
    long long total = (long long)Etot * H;
    long long i = (long long)blockIdx.x * blockDim.x + threadIdx.x;
    long long stride = (long long)gridDim.x * blockDim.x;
    for (; i < total; i += stride) {
        int hh = (int)(i % H);
        int e  = (int)(i / H);
        int s, d; edge_nodes(ei, E, e, s, d);
        float v = lrelu(asv[(size_t)s * H + hh] + adv[(size_t)d * H + hh]);
        atomicMaxFloat(&mx[(size_t)d * H + hh], v);
    }
}

// ---------------------------------------------------------------------------
// edge pass 2: p = exp(e - max[dst]); store p, accumulate segment sum
// ---------------------------------------------------------------------------
__global__ void edge_exp_kernel(const int* __restrict__ ei, int E, int Etot,
                                const float* __restrict__ asv, const float* __restrict__ adv,
                                const float* __restrict__ mx, float* __restrict__ ae,
                                float* __restrict__ sm, int H) {
    long long total = (long long)Etot * H;
    long long i = (long long)blockIdx.x * blockDim.x + threadIdx.x;
    long long stride = (long long)gridDim.x * blockDim.x;
    for (; i < total; i += stride) {
        int hh = (int)(i % H);
        int e  = (int)(i / H);
        int s, d; edge_nodes(ei, E, e, s, d);
        float v = lrelu(asv[(size_t)s * H + hh] + adv[(size_t)d * H + hh]);
        float p = expf(v - mx[(size_t)d * H + hh]);
        ae[i] = p;
        atomicAdd(&sm[(size_t)d * H + hh], p);
    }
}

// ---------------------------------------------------------------------------
// edge pass 3 (vector, C%4==0): agg[dst,h,:] += (p/sum) * h[src,h,:]
// ---------------------------------------------------------------------------
__global__ void edge_agg_vec_kernel(const int* __restrict__ ei, int E, int Etot,
                                    const float* __restrict__ ae, const float* __restrict__ sm,
                                    const float* __restrict__ h, float* __restrict__ agg,
                                    int H, int C) {
    int C4 = C >> 2;
    long long total = (long long)Etot * H * C4;
    long long i = (long long)blockIdx.x * blockDim.x + threadIdx.x;
    long long stride = (long long)gridDim.x * blockDim.x;
    for (; i < total; i += stride) {
        int c4 = (int)(i % C4);
        long long t = i / C4;
        int hh = (int)(t % H);
        int e  = (int)(t / H);
        int s, d; edge_nodes(ei, E, e, s, d);
        float coef = ae[(size_t)e * H + hh] / (sm[(size_t)d * H + hh] + 1e-16f);
        float4 hv = *(const float4*)(h + ((size_t)s * H + hh) * C + c4 * 4);
        float* ap = agg + ((size_t)d * H + hh) * C + c4 * 4;
        atomicAdd(ap + 0, coef * hv.x);
        atomicAdd(ap + 1, coef * hv.y);
        atomicAdd(ap + 2, coef * hv.z);
        atomicAdd(ap + 3, coef * hv.w);
    }
}

// scalar variant for small C (layer 4, C=2)
__global__ void edge_agg_scalar_kernel(const int* __restrict__ ei, int E, int Etot,
                                       const float* __restrict__ ae, const float* __restrict__ sm,
                                       const float* __restrict__ h, float* __restrict__ agg,
                                       int H, int C) {
    long long total = (long long)Etot * H * C;
    long long i = (long long)blockIdx.x * blockDim.x + threadIdx.x;
    long long stride = (long long)gridDim.x * blockDim.x;
    for (; i < total; i += stride) {
        int c = (int)(i % C);
        long long t = i / C;
        int hh = (int)(t % H);
        int e  = (int)(t / H);
        int s, d; edge_nodes(ei, E, e, s, d);
        float coef = ae[(size_t)e * H + hh] / (sm[(size_t)d * H + hh] + 1e-16f);
        atomicAdd(&agg[((size_t)d * H + hh) * C + c],
                  coef * h[((size_t)s * H + hh) * C + c]);
    }
}

// ---------------------------------------------------------------------------
// out = (elu?)(agg + b[f])
// ---------------------------------------------------------------------------
__global__ void finalize_kernel(const float* __restrict__ agg, const float* __restrict__ b,
                                float* __restrict__ out, long long total, int F, int do_elu) {
    long long i = (long long)blockIdx.x * blockDim.x + threadIdx.x;
    long long stride = (long long)gridDim.x * blockDim.x;
    for (; i < total; i += stride) {
        int f = (int)(i % F);
        float v = agg[i] + b[f];
        if (do_elu && v < 0.0f) v = expm1f(v);
        out[i] = v;
    }
}

// ---------------------------------------------------------------------------
// layer 4 fused transform: h4 = feat @ W4 (K x 2) and attention logits
// ---------------------------------------------------------------------------
__global__ void l4_transform_kernel(const float* __restrict__ feat, const float* __restrict__ W,
                                    const float* __restrict__ aS, const float* __restrict__ aD,
                                    float* __restrict__ h4, float* __restrict__ asv,
                                    float* __restrict__ adv, int Nn, int K) {
    int n = blockIdx.x * blockDim.x + threadIdx.x;
    if (n >= Nn) return;
    const float* fp = feat + (size_t)n * K;
    float a0 = 0.0f, a1 = 0.0f;
    for (int k = 0; k < K; ++k) {
        float xv = fp[k];
        a0 += xv * W[k * 2 + 0];
        a1 += xv * W[k * 2 + 1];
    }
    h4[(size_t)n * 2 + 0] = a0;
    h4[(size_t)n * 2 + 1] = a1;
    asv[n] = a0 * aS[0] + a1 * aS[1];
    adv[n] = a0 * aD[0] + a1 * aD[1];
}

// ---------------------------------------------------------------------------
// host
// ---------------------------------------------------------------------------
static inline int gs_blocks(long long total, int bs) {
    long long b = (total + bs - 1) / bs;
    if (b > (1LL << 20)) b = (1LL << 20);
    if (b < 1) b = 1;
    return (int)b;
}

extern "C" void kernel_launch(void* const* d_in, const int* in_sizes, int n_in,
                              void* d_out, int out_size, void* d_ws, size_t ws_size,
                              hipStream_t stream) {
    const float* x      = (const float*)d_in[0];
    const int*   ei     = (const int*)d_in[1];
    const float* W1     = (const float*)d_in[2];
    const float* asrc1  = (const float*)d_in[3];
    const float* adst1  = (const float*)d_in[4];
    const float* b1     = (const float*)d_in[5];
    const float* W2     = (const float*)d_in[6];
    const float* asrc2  = (const float*)d_in[7];
    const float* adst2  = (const float*)d_in[8];
    const float* b2     = (const float*)d_in[9];
    const float* W3     = (const float*)d_in[10];
    const float* asrc3  = (const float*)d_in[11];
    const float* adst3  = (const float*)d_in[12];
    const float* b3     = (const float*)d_in[13];
    const float* W4     = (const float*)d_in[14];
    const float* asrc4  = (const float*)d_in[15];
    const float* adst4  = (const float*)d_in[16];
    const float* b4     = (const float*)d_in[17];

    const int Nn   = in_sizes[0] / 32;   // 10000
    const int E    = in_sizes[1] / 2;    // 160000
    const int Etot = E + Nn;             // + self loops

    const float NEGINF = -__builtin_huge_valf();

    // workspace layout (floats)
    float* ws  = (float*)d_ws;
    float* P   = ws;                         // N*1024  agg / activated features
    float* Q   = P  + (size_t)Nn * 1024;     // N*1024  h (transformed features)
    float* asv = Q  + (size_t)Nn * 1024;     // N*8
    float* adv = asv + (size_t)Nn * 8;       // N*8
    float* mx  = adv + (size_t)Nn * 8;       // N*8
    float* sm  = mx  + (size_t)Nn * 8;       // N*8
    float* ae  = sm  + (size_t)Nn * 8;       // Etot*8

    const int BS = 256;

    // generic GAT layer (layers 1-3): h in Q, agg+output in P
    auto run_layer = [&](const float* xin, int Fin, const float* W, const float* aS,
                         const float* aD, const float* bias, int H, int C) {
        int Fout = H * C;
        long long waves = (long long)(Nn >> 4) * (Fout >> 6);   // 16x64 slabs
        gemm_wmma_f32<<<(int)((waves + 7) / 8), BS, 0, stream>>>(xin, W, Q, Nn, Fin, Fout);

        long long awaves = (long long)Nn * H;
        alpha_kernel<<<(int)((awaves + 7) / 8), BS, 0, stream>>>(Q, aS, aD, asv, adv, Nn, H, C);

        long long nh = (long long)Nn * H;
        fill_kernel<<<gs_blocks(nh, BS), BS, 0, stream>>>(mx, NEGINF, nh);
        fill_kernel<<<gs_blocks(nh, BS), BS, 0, stream>>>(sm, 0.0f, nh);
        long long nagg = (long long)Nn * Fout;
        fill_kernel<<<gs_blocks(nagg, BS), BS, 0, stream>>>(P, 0.0f, nagg);

        long long eh = (long long)Etot * H;
        edge_max_kernel<<<gs_blocks(eh, BS), BS, 0, stream>>>(ei, E, Etot, asv, adv, mx, H);
        edge_exp_kernel<<<gs_blocks(eh, BS), BS, 0, stream>>>(ei, E, Etot, asv, adv, mx, ae, sm, H);

        long long ework = eh * (C >> 2);
        edge_agg_vec_kernel<<<gs_blocks(ework, BS), BS, 0, stream>>>(ei, E, Etot, ae, sm, Q, P, H, C);

        finalize_kernel<<<gs_blocks(nagg, BS), BS, 0, stream>>>(P, bias, P, nagg, Fout, 1);
    };

    // layer 1: 32 -> 8 heads x 128 (concat)
    run_layer(x, 32, W1, asrc1, adst1, b1, 8, 128);
    // layer 2: 1024 -> 128
    run_layer(P, 1024, W2, asrc2, adst2, b2, 1, 128);
    // layer 3: 128 -> 128
    run_layer(P, 128, W3, asrc3, adst3, b3, 1, 128);

    // layer 4: 128 -> 2, no ELU, output to d_out
    {
        l4_transform_kernel<<<(Nn + BS - 1) / BS, BS, 0, stream>>>(
            P, W4, asrc4, adst4, Q, asv, adv, Nn, 128);

        long long nh = Nn;
        fill_kernel<<<gs_blocks(nh, BS), BS, 0, stream>>>(mx, NEGINF, nh);
        fill_kernel<<<gs_blocks(nh, BS), BS, 0, stream>>>(sm, 0.0f, nh);
        long long nagg = (long long)Nn * 2;
        fill_kernel<<<gs_blocks(nagg, BS), BS, 0, stream>>>(P, 0.0f, nagg);

        long long eh = Etot;
        edge_max_kernel<<<gs_blocks(eh, BS), BS, 0, stream>>>(ei, E, Etot, asv, adv, mx, 1);
        edge_exp_kernel<<<gs_blocks(eh, BS), BS, 0, stream>>>(ei, E, Etot, asv, adv, mx, ae, sm, 1);
        edge_agg_scalar_kernel<<<gs_blocks(eh * 2, BS), BS, 0, stream>>>(
            ei, E, Etot, ae, sm, Q, P, 1, 2);

        finalize_kernel<<<gs_blocks(nagg, BS), BS, 0, stream>>>(P, b4, (float*)d_out, nagg, 2, 0);
    }
}